// DgaRawSequence_1821066134063
// MI455X (gfx1250) — compile-verified
//
#include <hip/hip_runtime.h>

#define BATCH 128
#define SEQ   4096
#define NIN   6
#define NN    64
#define MOT   3
#define UNFOLDS 6
#define EPSV  1e-8f
#define LOG2E 1.4426950408889634f

typedef float v2f __attribute__((ext_vector_type(2)));
typedef float v8f __attribute__((ext_vector_type(8)));

// One block per batch element. 256 threads = 8 wave32.
// Wave w: j-tile = (w&3)*16, i-half = (w>>2)*32.
// Per ODE unfold, each wave computes 16(j) x 32(i) sigmoid activations and
// reduces over i with V_WMMA_F32_16X16X4_F32 against an all-ones B matrix
// (D[m,n] = sum_k A[m,k] + C[m,n] -> column sums land in the C/D layout).
__global__ __launch_bounds__(256) void ltc_scan_kernel(
    const float* __restrict__ x,
    const float* __restrict__ mean_us, const float* __restrict__ std_us,
    const float* __restrict__ input_w, const float* __restrict__ input_b,
    const float* __restrict__ output_w, const float* __restrict__ output_b,
    const float* __restrict__ gleak, const float* __restrict__ vleak,
    const float* __restrict__ cm,
    const float* __restrict__ sW, const float* __restrict__ smu,
    const float* __restrict__ ssig, const float* __restrict__ serev,
    const float* __restrict__ W, const float* __restrict__ mu,
    const float* __restrict__ sigma, const float* __restrict__ erev,
    float* __restrict__ out)
{
  extern __shared__ unsigned char smem[];
  float4* pk   = (float4*)smem;                       // 64KB: (sig*, mu*sig*, W, W*erev)
  float4* spk  = (float4*)(smem + 65536);             // 6KB sensory packed
  float*  vlds = (float*)(smem + 65536 + 6144);       // 64 neuron states
  float*  pnum = vlds + NN;                           // [2][64] partial numerators
  float*  pden = pnum + 2*NN;                         // [2][64] partial denominators
  float*  ulds = pden + 2*NN;                         // 6 mapped inputs

  const int tid   = threadIdx.x;
  const int b     = blockIdx.x;
  const int lane  = tid & 31;
  const int wv    = tid >> 5;
  const int jbase = (wv & 3) << 4;   // 0,16,32,48
  const int ihalf = wv >> 2;         // 0 or 1 (i in [0,32) or [32,64))
  const int khalf = lane >> 4;       // A-matrix K half per ISA layout
  const int j     = jbase + (lane & 15);

  // ---- one-time parameter packing into LDS ----
  for (int k = tid; k < NN*NN; k += 256) {
    float w = W[k], m_ = mu[k], sg = sigma[k], er = erev[k];
    pk[k] = make_float4(sg*LOG2E, m_*sg*LOG2E, w, w*er);
  }
  for (int k = tid; k < NIN*NN; k += 256) {
    float w = sW[k], m_ = smu[k], sg = ssig[k], er = serev[k];
    spk[k] = make_float4(sg*LOG2E, m_*sg*LOG2E, w, w*er);
  }
  if (tid < NN) vlds[tid] = 0.0f;

  // per-neuron constants held by updater threads (tid<64)
  float cmt = 0.f, gl = 0.f, gvl = 0.f;
  if (tid < NN) { cmt = cm[tid]*(float)UNFOLDS; gl = gleak[tid]; gvl = gl*vleak[tid]; }

  // input affine (normalize + input map folded): u = x*scl + bis
  float scl = 0.f, bis = 0.f, xn = 0.f;
  const float* xb = x + (size_t)b*SEQ*NIN;
  if (tid < NIN) {
    float invs = 1.0f/std_us[tid];
    scl = input_w[tid]*invs;
    bis = input_b[tid] - mean_us[tid]*scl;
    xn  = xb[tid];                       // prefetch step 0
  }
  // output affine (output map + denormalize folded)
  float ow2 = 0.f, ob2 = 0.f;
  if (tid < MOT) {
    ow2 = output_w[tid]*std_us[tid];
    ob2 = output_b[tid]*std_us[tid] + mean_us[tid];
  }
  float* outb = out + (size_t)b*SEQ*MOT;
  __syncthreads();

  const v2f ones = {1.0f, 1.0f};

  for (int s = 0; s < SEQ; ++s) {
    // publish mapped input for this step; prefetch next step's x
    if (tid < NIN) {
      ulds[tid] = __builtin_fmaf(xn, scl, bis);
      int s2 = (s + 1 < SEQ) ? (s + 1) : s;
      xn = xb[s2*NIN + tid];
    }
    __syncthreads();

    // sensory synapses (tiny: 6 x 64), kept in updater-thread registers
    float wns = 0.0f, wds = 0.0f;
    if (tid < NN) {
      #pragma unroll
      for (int in_ = 0; in_ < NIN; ++in_) {
        float u  = ulds[in_];
        float4 sp = spk[in_*NN + tid];
        float t  = __builtin_fmaf(-u, sp.x, sp.y);          // (mu-u)*sig*log2e
        float r  = __builtin_amdgcn_rcpf(1.0f + __builtin_amdgcn_exp2f(t));
        wns = __builtin_fmaf(sp.w, r, wns);
        wds = __builtin_fmaf(sp.z, r, wds);
      }
    }

    for (int uf = 0; uf < UNFOLDS; ++uf) {
      v8f cN = {}; v8f cD = {};
      #pragma unroll
      for (int c = 0; c < 8; ++c) {
        // A-matrix 16x4 f32 layout: lanes<16 hold K=0,1; lanes>=16 hold K=2,3
        int i0 = ihalf*32 + c*4 + khalf*2;
        float v0 = vlds[i0];
        float v1 = vlds[i0+1];
        float4 p0 = pk[i0*NN + j];
        float4 p1 = pk[(i0+1)*NN + j];
        float t0 = __builtin_fmaf(-v0, p0.x, p0.y);
        float t1 = __builtin_fmaf(-v1, p1.x, p1.y);
        float r0 = __builtin_amdgcn_rcpf(1.0f + __builtin_amdgcn_exp2f(t0));
        float r1 = __builtin_amdgcn_rcpf(1.0f + __builtin_amdgcn_exp2f(t1));
        v2f aN, aD;
        aN.x = p0.w*r0; aN.y = p1.w*r1;   // W*erev*sigmoid
        aD.x = p0.z*r0; aD.y = p1.z*r1;   // W*sigmoid
        // K-reduction on the matrix pipe: D = A x ones + C
        cN = __builtin_amdgcn_wmma_f32_16x16x4_f32(false, aN, false, ones,
                                                   (short)0, cN, false, false);
        cD = __builtin_amdgcn_wmma_f32_16x16x4_f32(false, aD, false, ones,
                                                   (short)0, cD, false, false);
      }
      // C/D layout: lane0 VGPR p = D[p, 0] (j=jbase+p); lane16 VGPR p = D[p+8, 0]
      if ((lane & 15) == 0) {
        int jb = jbase + khalf*8;
        #pragma unroll
        for (int p = 0; p < 8; ++p) {
          pnum[ihalf*NN + jb + p] = cN[p];
          pden[ihalf*NN + jb + p] = cD[p];
        }
      }
      __syncthreads();
      if (tid < NN) {
        float v    = vlds[tid];
        float numv = cmt*v + gvl + wns + pnum[tid] + pnum[NN + tid];
        float denv = cmt + gl + wds + pden[tid] + pden[NN + tid];
        vlds[tid]  = numv / (denv + EPSV);
      }
      __syncthreads();
    }

    if (tid < MOT) {
      outb[s*MOT + tid] = __builtin_fmaf(vlds[tid], ow2, ob2);
    }
  }
}

extern "C" void kernel_launch(void* const* d_in, const int* in_sizes, int n_in,
                              void* d_out, int out_size, void* d_ws, size_t ws_size,
                              hipStream_t stream) {
  const float* x        = (const float*)d_in[0];
  const float* mean_us  = (const float*)d_in[1];
  const float* std_us   = (const float*)d_in[2];
  const float* input_w  = (const float*)d_in[3];
  const float* input_b  = (const float*)d_in[4];
  const float* output_w = (const float*)d_in[5];
  const float* output_b = (const float*)d_in[6];
  const float* gleak    = (const float*)d_in[7];
  const float* vleak    = (const float*)d_in[8];
  const float* cm       = (const float*)d_in[9];
  const float* sW       = (const float*)d_in[10];
  const float* smu      = (const float*)d_in[11];
  const float* ssig     = (const float*)d_in[12];
  const float* serev    = (const float*)d_in[13];
  const float* W        = (const float*)d_in[14];
  const float* mu       = (const float*)d_in[15];
  const float* sigma    = (const float*)d_in[16];
  const float* erev     = (const float*)d_in[17];
  float* out = (float*)d_out;

  // LDS: 64KB pk + 6KB spk + v(256B) + pnum(512B) + pden(512B) + ulds(32B)
  size_t shmem = 65536 + 6144 + NN*4 + 2*NN*4 + 2*NN*4 + 8*4;

  ltc_scan_kernel<<<dim3(BATCH), dim3(256), shmem, stream>>>(
      x, mean_us, std_us, input_w, input_b, output_w, output_b,
      gleak, vleak, cm, sW, smu, ssig, serev, W, mu, sigma, erev, out);
}